// MipRayMarcher2_43739946943221
// MI455X (gfx1250) — compile-verified
//
#include <hip/hip_runtime.h>
#include <math.h>
#include <stdint.h>

// Shapes fixed by the reference: (B,R,S,C)=(2,16384,48,32), Cd=64.
#define S_SAMP 48
#define NSEG   47
#define C_RGB  32
#define C_DINO 64
#define CBYTES (S_SAMP * C_RGB * 4)                      // 6144 B colors per ray
#define WAVES_PER_BLOCK 2
#define WAVE_FLOATS (S_SAMP * C_RGB + S_SAMP * C_DINO)   // 4608 floats = 18 KB per ray

// ---------------------------------------------------------------- init ------
__global__ void init_ws_kernel(unsigned* ws) {
    ws[0] = 0x7F800000u;  // running min (depths > 0 -> uint order == float order)
    ws[1] = 0x00000000u;  // running max
    ws[2] = 0x00000000u;  // variance accumulator (float 0.0)
}

// ------------------------------------------------------- depth min / max ----
__global__ void depth_minmax_kernel(const float* __restrict__ depths, int n,
                                    unsigned* __restrict__ ws) {
    unsigned vmin = 0x7F800000u, vmax = 0u;
    for (int i = blockIdx.x * blockDim.x + threadIdx.x; i < n;
         i += gridDim.x * blockDim.x) {
        unsigned b = __float_as_uint(depths[i]);  // positive floats only
        vmin = min(vmin, b);
        vmax = max(vmax, b);
    }
#pragma unroll
    for (int off = 16; off >= 1; off >>= 1) {
        vmin = min(vmin, (unsigned)__shfl_xor((int)vmin, off));
        vmax = max(vmax, (unsigned)__shfl_xor((int)vmax, off));
    }
    if ((threadIdx.x & 31) == 0) {
        atomicMin(&ws[0], vmin);
        atomicMax(&ws[1], vmax);
    }
}

// ------------------------------------------------------------- marcher ------
// One wave32 per ray.
//  * Colors+dinos streamed global->LDS with the CDNA5 async DMA path
//    (ASYNCcnt). Async loads retire IN ORDER, so issuing in segment-group
//    order lets partial s_wait_asynccnt fences gate the consumer loop while
//    the tail of the 18 KB stream is still in flight.
//  * Transmittance chain is computed LANE-PARALLEL (one segment per lane,
//    2 batches), with a Hillis-Steele multiplicative scan for the cumprod
//    and a shfl_xor wave reduction for the depth moments -- 2 transcendental
//    evaluations per ray instead of 47, fully overlapped with the DMA.

#define ASYNC_LDS_B128_NT(loff, gptr)                                          \
    asm volatile("global_load_async_to_lds_b128 %0, %1, off th:TH_LOAD_NT"     \
                 :: "v"(loff), "v"(gptr) : "memory")

#define WAIT_ASYNC(n) asm volatile("s_wait_asynccnt " #n ::: "memory")

__launch_bounds__(WAVES_PER_BLOCK * 32)
__global__ void march_kernel(const float* __restrict__ colors,
                             const float* __restrict__ densities,
                             const float* __restrict__ depths,
                             const float* __restrict__ dinos,
                             float* __restrict__ out_rgb,
                             float* __restrict__ out_dep,
                             float* __restrict__ out_w,
                             float* __restrict__ out_bg,
                             float* __restrict__ out_dino,
                             float* __restrict__ out_wt,
                             float* __restrict__ var_acc,
                             int n_rays, float inv_nrays) {
    __shared__ float smem[WAVES_PER_BLOCK * WAVE_FLOATS];
    const int wave = threadIdx.x >> 5;
    const int lane = threadIdx.x & 31;
    const int ray  = blockIdx.x * WAVES_PER_BLOCK + wave;
    if (ray >= n_rays) return;

    float* sw = &smem[wave * WAVE_FLOATS];
    const float* cg = colors + (size_t)ray * (S_SAMP * C_RGB);
    const float* dg = dinos  + (size_t)ray * (S_SAMP * C_DINO);

    // Low 32 bits of the generic shared-aperture address == LDS byte offset.
    uint32_t lbase = (uint32_t)(uintptr_t)sw;

    // Issue 12 groups, each covering 4 segments:
    //   1x colors chunk (512 B) + 2x dinos chunks (2x512 B), 3 DMA ops/group.
    // After group G retires: segments 0..4G+3 are resident in LDS.
#pragma unroll
    for (int g = 0; g < 12; ++g) {
        const float* gc = cg + g * 128 + lane * 4;
        uint32_t lc = lbase + (uint32_t)g * 512u + (uint32_t)lane * 16u;
        ASYNC_LDS_B128_NT(lc, gc);
        const float* gd0 = dg + g * 256 + lane * 4;
        uint32_t ld0 = lbase + (uint32_t)CBYTES + (uint32_t)g * 1024u
                     + (uint32_t)lane * 16u;
        ASYNC_LDS_B128_NT(ld0, gd0);
        const float* gd1 = dg + g * 256 + 128 + lane * 4;
        uint32_t ld1 = lbase + (uint32_t)CBYTES + (uint32_t)g * 1024u + 512u
                     + (uint32_t)lane * 16u;
        ASYNC_LDS_B128_NT(ld1, gd1);
    }

    // ---- lane-parallel transmittance chain (overlaps the DMA) -------------
    const float* dep = depths    + (size_t)ray * S_SAMP;
    const float* den = densities + (size_t)ray * S_SAMP;
    const bool lb = (lane < NSEG - 32);         // lanes 0..14 own segs 32..46
    float depA0 = dep[lane],     depA1 = dep[lane + 1];
    float denA0 = den[lane],     denA1 = den[lane + 1];
    float depB0 = lb ? dep[32 + lane] : 0.0f;
    float depB1 = lb ? dep[33 + lane] : 0.0f;
    float denB0 = lb ? den[32 + lane] : 0.0f;
    float denB1 = lb ? den[33 + lane] : 0.0f;

    // batch A: segment s = lane (0..31)
    float deltaA = depA1 - depA0;
    float dmidA  = 0.5f * (depA0 + depA1);
    float xA  = 0.5f * (denA0 + denA1) - 1.0f;
    // jax.nn.softplus == logaddexp(x,0) == max(x,0)+log1p(exp(-|x|))
    float spA = fmaxf(xA, 0.0f) + log1pf(expf(-fabsf(xA)));
    float alphaA = 1.0f - expf(-spA * deltaA);
    float aA = 1.0f - alphaA + 1e-10f;

    // batch B: segment s = 32 + lane (lanes 0..14); identity elsewhere
    float deltaB = depB1 - depB0;
    float dmidB  = 0.5f * (depB0 + depB1);
    float xB  = 0.5f * (denB0 + denB1) - 1.0f;
    float spB = fmaxf(xB, 0.0f) + log1pf(expf(-fabsf(xB)));
    float alphaB = lb ? (1.0f - expf(-spB * deltaB)) : 0.0f;
    float aB = lb ? (1.0f - alphaB + 1e-10f) : 1.0f;

    // inclusive multiplicative scans (Hillis-Steele, 5 shfl_up steps each)
    float pA = aA;
#pragma unroll
    for (int off = 1; off < 32; off <<= 1) {
        float q = __shfl_up(pA, off);
        if (lane >= off) pA *= q;
    }
    float TA = __shfl_up(pA, 1);                // exclusive: T_s for s=lane
    if (lane == 0) TA = 1.0f;
    float PAtot = __shfl(pA, 31);               // prod over segments 0..31

    float pB = aB;
#pragma unroll
    for (int off = 1; off < 32; off <<= 1) {
        float q = __shfl_up(pB, off);
        if (lane >= off) pB *= q;
    }
    float TB = __shfl_up(pB, 1);
    if (lane == 0) TB = 1.0f;
    TB *= PAtot;                                // exclusive T for s = 32+lane

    float wA = alphaA * TA;                     // weight of segment lane
    float wB = alphaB * TB;                     // weight of segment 32+lane (0 if !lb)
    float bg = __shfl(TB, NSEG - 33);           // T_46 = bg_lambda

    // depth moments: S0 = sum w, S1 = sum w*dmid, S2 = sum w*dmid^2
    float m0 = wA + wB;
    float m1 = wA * dmidA + wB * dmidB;
    float m2 = wA * dmidA * dmidA + wB * dmidB * dmidB;
#pragma unroll
    for (int off = 16; off >= 1; off >>= 1) {
        m0 += __shfl_xor(m0, off);
        m1 += __shfl_xor(m1, off);
        m2 += __shfl_xor(m2, off);
    }

    // ---- channel accumulation: consume LDS behind the in-order DMA --------
    const float* sc = sw;                       // [48][32] colors
    const float* sd = sw + S_SAMP * C_RGB;      // [48][64] dinos
    float crgb = 0.0f, dinA = 0.0f, dinB = 0.0f;

    WAIT_ASYNC(33);                             // group 0 resident (segs 0..3)
    float cprev = sc[lane];
    float dApv  = sd[2 * lane];
    float dBpv  = sd[2 * lane + 1];

#define MARCH_STEP(i)                                                          \
    do {                                                                       \
        const int j = (i) + 1;                                                 \
        float w = ((i) < 32) ? __shfl(wA, (i)) : __shfl(wB, (i) - 32);         \
        float cn  = sc[j * C_RGB + lane];                                      \
        float dAn = sd[j * C_DINO + 2 * lane];                                 \
        float dBn = sd[j * C_DINO + 2 * lane + 1];                             \
        crgb += w * 0.5f * (cprev + cn);                                       \
        dinA += w * 0.5f * (dApv + dAn);                                       \
        dinB += w * 0.5f * (dBpv + dBn);                                       \
        cprev = cn; dApv = dAn; dBpv = dBn;                                    \
    } while (0)

    /* group G resident after s_wait_asynccnt (33-3G); steps i <= 4G+2 */
    MARCH_STEP(0);  MARCH_STEP(1);  MARCH_STEP(2);
    WAIT_ASYNC(30); MARCH_STEP(3);  MARCH_STEP(4);  MARCH_STEP(5);  MARCH_STEP(6);
    WAIT_ASYNC(27); MARCH_STEP(7);  MARCH_STEP(8);  MARCH_STEP(9);  MARCH_STEP(10);
    WAIT_ASYNC(24); MARCH_STEP(11); MARCH_STEP(12); MARCH_STEP(13); MARCH_STEP(14);
    WAIT_ASYNC(21); MARCH_STEP(15); MARCH_STEP(16); MARCH_STEP(17); MARCH_STEP(18);
    WAIT_ASYNC(18); MARCH_STEP(19); MARCH_STEP(20); MARCH_STEP(21); MARCH_STEP(22);
    WAIT_ASYNC(15); MARCH_STEP(23); MARCH_STEP(24); MARCH_STEP(25); MARCH_STEP(26);
    WAIT_ASYNC(12); MARCH_STEP(27); MARCH_STEP(28); MARCH_STEP(29); MARCH_STEP(30);
    WAIT_ASYNC(9);  MARCH_STEP(31); MARCH_STEP(32); MARCH_STEP(33); MARCH_STEP(34);
    WAIT_ASYNC(6);  MARCH_STEP(35); MARCH_STEP(36); MARCH_STEP(37); MARCH_STEP(38);
    WAIT_ASYNC(3);  MARCH_STEP(39); MARCH_STEP(40); MARCH_STEP(41); MARCH_STEP(42);
    WAIT_ASYNC(0);  MARCH_STEP(43); MARCH_STEP(44); MARCH_STEP(45); MARCH_STEP(46);
#undef MARCH_STEP

    // ---- outputs (write-once -> non-temporal) -----------------------------
    __builtin_nontemporal_store(crgb * 2.0f - 1.0f,
                                &out_rgb[(size_t)ray * C_RGB + lane]);
    __builtin_nontemporal_store((dinA + 1.0f - m0) * 2.0f - 1.0f,
                                &out_dino[(size_t)ray * C_DINO + 2 * lane]);
    __builtin_nontemporal_store((dinB + 1.0f - m0) * 2.0f - 1.0f,
                                &out_dino[(size_t)ray * C_DINO + 2 * lane + 1]);
    __builtin_nontemporal_store(wA, &out_w[(size_t)ray * NSEG + lane]);
    if (lb)
        __builtin_nontemporal_store(wB, &out_w[(size_t)ray * NSEG + 32 + lane]);
    if (lane == 0) {
        __builtin_nontemporal_store(m0, &out_wt[ray]);
        __builtin_nontemporal_store(bg, &out_bg[ray]);
        out_dep[ray] = m1 / m0;   // regular store: finalize pass re-reads it
        // sum w*(d - S1)^2 with UNNORMALIZED mean S1 (matches reference):
        float varn = m2 - 2.0f * m1 * m1 + m1 * m1 * m0;
        atomicAdd(var_acc, (varn / (m0 + 1e-6f)) * inv_nrays);
    }
}

// ------------------------------------------------------------ finalize ------
__global__ void finalize_kernel(float* __restrict__ out_dep,
                                const unsigned* __restrict__ ws,
                                float* __restrict__ out_var, int n_rays) {
    int i = blockIdx.x * blockDim.x + threadIdx.x;
    float mn = __uint_as_float(ws[0]);
    float mx = __uint_as_float(ws[1]);
    if (i < n_rays) {
        float d = out_dep[i];
        if (d != d) d = __builtin_inff();       // nan_to_num(nan=inf)
        d = fminf(fmaxf(d, mn), mx);            // clip to [min(depths), max(depths)]
        out_dep[i] = d;
    }
    if (i == 0) *out_var = __uint_as_float(ws[2]);
}

// -------------------------------------------------------------- launch ------
extern "C" void kernel_launch(void* const* d_in, const int* in_sizes, int n_in,
                              void* d_out, int out_size, void* d_ws, size_t ws_size,
                              hipStream_t stream) {
    const float* colors    = (const float*)d_in[0];
    const float* densities = (const float*)d_in[1];
    const float* depths    = (const float*)d_in[2];
    const float* dinos     = (const float*)d_in[3];

    const int n_depth = in_sizes[2];            // B*R*S
    const int n_rays  = n_depth / S_SAMP;       // B*R = 32768

    float* out = (float*)d_out;
    size_t off = 0;
    float* out_rgb  = out + off; off += (size_t)n_rays * C_RGB;
    float* out_dep  = out + off; off += (size_t)n_rays;
    float* out_w    = out + off; off += (size_t)n_rays * NSEG;
    float* out_bg   = out + off; off += (size_t)n_rays;
    float* out_dino = out + off; off += (size_t)n_rays * C_DINO;
    float* out_wt   = out + off; off += (size_t)n_rays;
    float* out_var  = out + off;

    unsigned* ws = (unsigned*)d_ws;

    init_ws_kernel<<<1, 1, 0, stream>>>(ws);
    depth_minmax_kernel<<<256, 256, 0, stream>>>(depths, n_depth, ws);

    const int blocks = (n_rays + WAVES_PER_BLOCK - 1) / WAVES_PER_BLOCK;
    march_kernel<<<blocks, WAVES_PER_BLOCK * 32, 0, stream>>>(
        colors, densities, depths, dinos,
        out_rgb, out_dep, out_w, out_bg, out_dino, out_wt,
        (float*)(ws + 2), n_rays, 1.0f / (float)n_rays);

    finalize_kernel<<<(n_rays + 255) / 256, 256, 0, stream>>>(out_dep, ws, out_var, n_rays);
}